// SparseMoE_39067022525073
// MI455X (gfx1250) — compile-verified
//
#include <hip/hip_runtime.h>
#include <math.h>

#define T_TOKENS 65536
#define H 128
#define E 8
#define F 256
#define ROWS 32
#define MAXSLOTS (2*T_TOKENS + E*ROWS)   // 131072 + 256 = 131328
#define NBLK_GEMM (MAXSLOTS/ROWS)        // 4104
#define LN_EPS 1e-5f
#define XS_STRIDE 132                    // 128 + 4 pad -> conflict-free ds reads
#define HS_STRIDE 260                    // 256 + 4 pad

typedef __attribute__((ext_vector_type(2))) float v2f;
typedef __attribute__((ext_vector_type(8))) float v8f;

static __device__ __forceinline__ v8f wmma_f32_16x16x4(v2f a, v2f b, v8f c) {
  return __builtin_amdgcn_wmma_f32_16x16x4_f32(false, a, false, b, (short)0, c,
                                               false, false);
}

// ---------------- Kernel 1: gating (one wave32 per token) ----------------
__global__ __launch_bounds__(256) void moe_gate(
    const float* __restrict__ x, const float* __restrict__ gumbel,
    const float* __restrict__ gW, const float* __restrict__ gb,
    int* __restrict__ top2e, float* __restrict__ top2w, int* __restrict__ count) {
  const int wave = threadIdx.x >> 5;
  const int lane = threadIdx.x & 31;
  const int t = blockIdx.x * 8 + wave;

  float4 xv = *(const float4*)(x + (size_t)t * H + lane * 4);
  float p[E];
#pragma unroll
  for (int e = 0; e < E; e++) p[e] = 0.f;
#pragma unroll
  for (int j = 0; j < 4; j++) {
    const int h = lane * 4 + j;
    const float xs = ((const float*)&xv)[j];
    float4 g0 = *(const float4*)(gW + (size_t)h * E);
    float4 g1 = *(const float4*)(gW + (size_t)h * E + 4);
    p[0] += xs * g0.x; p[1] += xs * g0.y; p[2] += xs * g0.z; p[3] += xs * g0.w;
    p[4] += xs * g1.x; p[5] += xs * g1.y; p[6] += xs * g1.z; p[7] += xs * g1.w;
  }
#pragma unroll
  for (int off = 16; off >= 1; off >>= 1) {
#pragma unroll
    for (int e = 0; e < E; e++) p[e] += __shfl_xor(p[e], off, 32);
  }
  if (lane == 0) {
    float v[E];
#pragma unroll
    for (int e = 0; e < E; e++)
      v[e] = (p[e] + gb[e] + gumbel[(size_t)t * E + e]) * 1.0f;  // TAU = 1
    int i0 = 0;
#pragma unroll
    for (int e = 1; e < E; e++) if (v[e] > v[i0]) i0 = e;         // first-max tiebreak
    int i1 = (i0 == 0) ? 1 : 0;
#pragma unroll
    for (int e = 0; e < E; e++)
      if (e != i0 && v[e] > v[i1]) i1 = e;
    const float m = fmaxf(v[i0], v[i1]);
    const float e0 = expf(v[i0] - m), e1 = expf(v[i1] - m);
    const float inv = 1.0f / (e0 + e1);
    top2e[2 * t] = i0;       top2e[2 * t + 1] = i1;
    top2w[2 * t] = e0 * inv; top2w[2 * t + 1] = e1 * inv;
    atomicAdd(&count[i0], 1);
    atomicAdd(&count[i1], 1);
  }
}

// ---------------- Kernel 2: padded segment offsets + pad-slot init ----------------
__global__ void moe_offsets(const int* __restrict__ count, int* __restrict__ off,
                            int* __restrict__ perm, float* __restrict__ slotw) {
  if (threadIdx.x == 0 && blockIdx.x == 0) {
    int acc = 0;
    for (int e = 0; e < E; e++) {
      off[e] = acc;
      const int c = count[e];
      const int pc = (c + ROWS - 1) & ~(ROWS - 1);
      for (int i = c; i < pc; i++) { perm[acc + i] = 0; slotw[acc + i] = 0.f; }
      acc += pc;
    }
    off[E] = acc;
  }
}

// ---------------- Kernel 3: scatter token assignments to slots ----------------
__global__ __launch_bounds__(256) void moe_scatter(
    const int* __restrict__ top2e, const float* __restrict__ top2w,
    const int* __restrict__ off, int* __restrict__ cursor,
    int* __restrict__ perm, float* __restrict__ slotw, int* __restrict__ slotid) {
  const int t = blockIdx.x * blockDim.x + threadIdx.x;
#pragma unroll
  for (int k = 0; k < 2; k++) {
    const int e = top2e[2 * t + k];
    const int p = off[e] + atomicAdd(&cursor[e], 1);
    perm[p] = t;
    slotw[p] = top2w[2 * t + k];
    slotid[2 * t + k] = p;
  }
}

// ---------------- Kernel 4: grouped expert GEMM (fp32 WMMA) ----------------
__global__ __launch_bounds__(256) void moe_expert_gemm(
    const float* __restrict__ x, const float* __restrict__ W1,
    const float* __restrict__ b1, const float* __restrict__ W2,
    const float* __restrict__ b2, const int* __restrict__ off,
    const int* __restrict__ perm, const float* __restrict__ slotw,
    float* __restrict__ slotOut) {
  __shared__ __align__(16) float Xs[ROWS * XS_STRIDE];
  __shared__ __align__(16) float Hs[ROWS * HS_STRIDE];
  __shared__ float Ws[ROWS];

  const int row0 = blockIdx.x * ROWS;
  if (row0 >= off[E]) return;  // uniform exit before any barrier/WMMA

  int e = 0;
#pragma unroll
  for (int i = 1; i < E; i++) if (row0 >= off[i]) e = i;
  const float* __restrict__ W1e = W1 + (size_t)e * H * F;
  const float* __restrict__ b1e = b1 + (size_t)e * F;
  const float* __restrict__ W2e = W2 + (size_t)e * F * H;
  const float* __restrict__ b2e = b2 + (size_t)e * H;

  const int wave = threadIdx.x >> 5;
  const int lane = threadIdx.x & 31;
  const int half = lane >> 4;
  const int l16 = lane & 15;

  // gather 32 token rows of X into LDS (float4, coalesced)
#pragma unroll
  for (int rr = 0; rr < 4; rr++) {
    const int r = wave * 4 + rr;
    const int tok = perm[row0 + r];
    float4 xv = *(const float4*)(x + (size_t)tok * H + lane * 4);
    *(float4*)(Xs + r * XS_STRIDE + lane * 4) = xv;
  }
  if (threadIdx.x < ROWS) Ws[threadIdx.x] = slotw[row0 + threadIdx.x];
  __syncthreads();

  // GEMM1: Hs = relu(X @ W1e + b1e)  [32 x 256]
  for (int ct2 = 0; ct2 < 2; ct2++) {
    const int n0 = (wave * 2 + ct2) * 16;
    v8f acc0 = {}; v8f acc1 = {};
    for (int k0 = 0; k0 < H; k0 += 4) {
      const int kk = k0 + 2 * half;
      v2f b;
      b.x = W1e[(size_t)kk * F + n0 + l16];
      b.y = W1e[(size_t)(kk + 1) * F + n0 + l16];
      v2f a0 = *(const v2f*)(Xs + l16 * XS_STRIDE + kk);
      v2f a1 = *(const v2f*)(Xs + (16 + l16) * XS_STRIDE + kk);
      acc0 = wmma_f32_16x16x4(a0, b, acc0);
      acc1 = wmma_f32_16x16x4(a1, b, acc1);
    }
    const float bias = b1e[n0 + l16];
#pragma unroll
    for (int r = 0; r < 8; r++) {
      const int m = r + 8 * half;
      Hs[m * HS_STRIDE + n0 + l16] = fmaxf(acc0[r] + bias, 0.f);
      Hs[(16 + m) * HS_STRIDE + n0 + l16] = fmaxf(acc1[r] + bias, 0.f);
    }
  }
  __syncthreads();

  // GEMM2: slotOut = slotw * (Hs @ W2e + b2e)  [32 x 128]
  {
    const int n0 = wave * 16;
    v8f acc0 = {}; v8f acc1 = {};
    for (int k0 = 0; k0 < F; k0 += 4) {
      const int kk = k0 + 2 * half;
      v2f b;
      b.x = W2e[(size_t)kk * H + n0 + l16];
      b.y = W2e[(size_t)(kk + 1) * H + n0 + l16];
      v2f a0 = *(const v2f*)(Hs + l16 * HS_STRIDE + kk);
      v2f a1 = *(const v2f*)(Hs + (16 + l16) * HS_STRIDE + kk);
      acc0 = wmma_f32_16x16x4(a0, b, acc0);
      acc1 = wmma_f32_16x16x4(a1, b, acc1);
    }
    const float bias = b2e[n0 + l16];
#pragma unroll
    for (int r = 0; r < 8; r++) {
      const int m = r + 8 * half;
      slotOut[(size_t)(row0 + m) * H + n0 + l16] = Ws[m] * (acc0[r] + bias);
      slotOut[(size_t)(row0 + 16 + m) * H + n0 + l16] = Ws[16 + m] * (acc1[r] + bias);
    }
  }
}

// ---------------- Kernel 5: gather + residual + LayerNorm ----------------
__global__ __launch_bounds__(256) void moe_finalize(
    const float* __restrict__ x, const float* __restrict__ slotOut,
    const int* __restrict__ slotid, const float* __restrict__ ln_g,
    const float* __restrict__ ln_b, float* __restrict__ out) {
  const int wave = threadIdx.x >> 5;
  const int lane = threadIdx.x & 31;
  const int t = blockIdx.x * 8 + wave;

  float4 xv = *(const float4*)(x + (size_t)t * H + lane * 4);
  const int s0 = slotid[2 * t], s1 = slotid[2 * t + 1];
  float4 o0 = *(const float4*)(slotOut + (size_t)s0 * H + lane * 4);
  float4 o1 = *(const float4*)(slotOut + (size_t)s1 * H + lane * 4);

  float y[4];
  y[0] = xv.x + o0.x + o1.x;
  y[1] = xv.y + o0.y + o1.y;
  y[2] = xv.z + o0.z + o1.z;
  y[3] = xv.w + o0.w + o1.w;

  float s = y[0] + y[1] + y[2] + y[3];
#pragma unroll
  for (int off = 16; off >= 1; off >>= 1) s += __shfl_xor(s, off, 32);
  const float mu = s * (1.0f / H);

  float sq = 0.f;
#pragma unroll
  for (int j = 0; j < 4; j++) { y[j] -= mu; sq += y[j] * y[j]; }
#pragma unroll
  for (int off = 16; off >= 1; off >>= 1) sq += __shfl_xor(sq, off, 32);
  const float rs = rsqrtf(sq * (1.0f / H) + LN_EPS);

  float4 g = *(const float4*)(ln_g + lane * 4);
  float4 bb = *(const float4*)(ln_b + lane * 4);
  float4 r;
  r.x = y[0] * rs * g.x + bb.x;
  r.y = y[1] * rs * g.y + bb.y;
  r.z = y[2] * rs * g.z + bb.z;
  r.w = y[3] * rs * g.w + bb.w;
  *(float4*)(out + (size_t)t * H + lane * 4) = r;
}

// ---------------- host launcher ----------------
extern "C" void kernel_launch(void* const* d_in, const int* in_sizes, int n_in,
                              void* d_out, int out_size, void* d_ws, size_t ws_size,
                              hipStream_t stream) {
  const float* x      = (const float*)d_in[0];
  const float* gumbel = (const float*)d_in[1];
  const float* gW     = (const float*)d_in[2];
  const float* gb     = (const float*)d_in[3];
  const float* W1     = (const float*)d_in[4];
  const float* b1     = (const float*)d_in[5];
  const float* W2     = (const float*)d_in[6];
  const float* b2     = (const float*)d_in[7];
  const float* lng    = (const float*)d_in[8];
  const float* lnb    = (const float*)d_in[9];
  float* out = (float*)d_out;

  char* w = (char*)d_ws;
  int*   count  = (int*)w;                                  // 8
  int*   cursor = count + 8;                                // 8
  int*   off    = cursor + 8;                               // 9
  size_t pos = 128;
  int*   top2e  = (int*)(w + pos);   pos += (size_t)2 * T_TOKENS * 4;
  float* top2w  = (float*)(w + pos); pos += (size_t)2 * T_TOKENS * 4;
  int*   slotid = (int*)(w + pos);   pos += (size_t)2 * T_TOKENS * 4;
  int*   perm   = (int*)(w + pos);   pos += (size_t)MAXSLOTS * 4;
  float* slotw  = (float*)(w + pos); pos += (size_t)MAXSLOTS * 4;
  float* slotOut = (float*)(w + pos);  // MAXSLOTS * H * 4 bytes

  hipMemsetAsync(count, 0, 16 * sizeof(int), stream);  // counts + cursors

  moe_gate<<<T_TOKENS / 8, 256, 0, stream>>>(x, gumbel, gW, gb, top2e, top2w, count);
  moe_offsets<<<1, 1, 0, stream>>>(count, off, perm, slotw);
  moe_scatter<<<T_TOKENS / 256, 256, 0, stream>>>(top2e, top2w, off, cursor, perm,
                                                  slotw, slotid);
  moe_expert_gemm<<<NBLK_GEMM, 256, 0, stream>>>(x, W1, b1, W2, b2, off, perm,
                                                 slotw, slotOut);
  moe_finalize<<<T_TOKENS / 8, 256, 0, stream>>>(x, slotOut, slotid, lng, lnb, out);
}